// HungarianMatcher_65962107732230
// MI455X (gfx1250) — compile-verified
//
#include <hip/hip_runtime.h>
#include <stdint.h>

#define BS_N 32
#define Q_N 1024
#define K_N 96
#define NT 256
#define CPT 4              // columns per thread = Q_N / NT
#define INF_F 1000000000.0f

#if defined(__HIP_DEVICE_COMPILE__) && defined(__gfx1250__)
  #define CDNA5_ASM 1
#else
  #define CDNA5_ASM 0
#endif

#if CDNA5_ASM && __has_builtin(__builtin_amdgcn_tensor_load_to_lds)
  #define HAVE_TDM 1
#else
  #define HAVE_TDM 0
#endif

typedef unsigned int u32x4 __attribute__((ext_vector_type(4)));
typedef int          i32x8 __attribute__((ext_vector_type(8)));
typedef int          i32x4 __attribute__((ext_vector_type(4)));

__device__ __forceinline__ uint32_t lds_byte_addr(const void* p) {
  // addrspacecast(AS3 -> flat) yields {shared_aperture_hi32, lds_offset32};
  // the low 32 bits are the LDS byte address the TDM/async ops expect.
  return (uint32_t)(uintptr_t)p;
}

__device__ __forceinline__ void argmin2(float& bv, int& bj, float v, int j) {
  if (v < bv || (v == bv && j < bj)) { bv = v; bj = j; }
}

__global__ __launch_bounds__(NT)
void hungarian_kernel(const float* __restrict__ logits,
                      const float* __restrict__ labels,
                      float* __restrict__ out) {
  __shared__ __align__(16) float s_rawlog[Q_N * 3];  // staged logits[b]   (12 KB)
  __shared__ __align__(16) float s_rawlab[384];      // staged labels[b][0:128) (1.5 KB)
  // packed per-column state: x=minv(A[j], Dm for used), y=way(bits), z=used(0/1), w=v
  __shared__ __align__(16) float4 s_colv[Q_N + 1];   // 16.4 KB, one ds_load_b128/column
  // packed per-query data: x=bx, y=by, z=sigmoid, w=pad
  __shared__ __align__(16) float4 s_qd[Q_N];         // 16 KB
  __shared__ float s_tx[K_N], s_ty[K_N];
  __shared__ float s_u[K_N + 1];
  __shared__ int   s_p[Q_N + 1];
  __shared__ int   s_a[K_N];
  __shared__ float s_redv[NT / 32];
  __shared__ int   s_redj[NT / 32];

  const int tid = threadIdx.x;
  const int b   = blockIdx.x;
  const float* lg = logits + (size_t)b * Q_N * 3;
  const float* lb = labels + (size_t)b * Q_N * 3;

  // -------- stage per-batch inputs into LDS (CDNA5 async / TDM paths) --------
#if CDNA5_ASM
  if (tid < 32) {
    // labels rows 0..95 (288 floats, padded to 384) via async global->LDS
    uint32_t labase = lds_byte_addr(s_rawlab);
    uint64_t gb = (uint64_t)(uintptr_t)lb;
    for (int t = 0; t < 3; ++t) {
      uint32_t off = (uint32_t)(t * 32 + tid) * 16u;
      uint32_t la = labase + off;
      uint64_t ga = gb + off;
      asm volatile("global_load_async_to_lds_b128 %0, %1, off"
                   :: "v"(la), "v"(ga) : "memory");
    }
#if HAVE_TDM
    if (tid == 0) {
      // Tensor Data Mover: 1-D tile of 3072 f32 elements -> LDS
      uint64_t ga = (uint64_t)(uintptr_t)lg;
      uint32_t la = lds_byte_addr(s_rawlog);
      const uint32_t n = Q_N * 3;
      u32x4 g0;
      g0.x = 1u;                                                   // count=1 (user D#)
      g0.y = la;                                                   // lds_addr
      g0.z = (uint32_t)ga;                                         // global_addr[31:0]
      g0.w = ((uint32_t)(ga >> 32) & 0x01FFFFFFu) | (2u << 30);    // addr[56:32] | type=2
      i32x8 g1;
      g1[0] = (int)(2u << 16);                                     // data_size = 4B
      g1[1] = (int)((n & 0xFFFFu) << 16);                          // tensor_dim0[15:0]
      g1[2] = (int)(((n >> 16) & 0xFFFFu) | (1u << 16));           // dim0[31:16] | tensor_dim1=1
      g1[3] = (int)((n & 0xFFFFu) << 16);                          // dim1[31:16]=0 | tile_dim0=n
      g1[4] = 0;                                                   // tile_dim1=0, tile_dim2=0
      g1[5] = (int)n;                                              // tensor_dim0_stride[31:0]
      g1[6] = 0;
      g1[7] = 0;
      i32x4 gz4 = (i32x4){0, 0, 0, 0};                             // groups 2/3 unused (<=2D)
      i32x8 gz8 = (i32x8){0, 0, 0, 0, 0, 0, 0, 0};
      __builtin_amdgcn_tensor_load_to_lds(g0, g1, gz4, gz4, gz8, 0);
    }
#else
    // no TDM builtin on this toolchain: async-copy the logits tile too
    uint32_t lgbase = lds_byte_addr(s_rawlog);
    uint64_t gl = (uint64_t)(uintptr_t)lg;
    for (int t = 0; t < 24; ++t) {
      uint32_t off = (uint32_t)(t * 32 + tid) * 16u;
      uint32_t la = lgbase + off;
      uint64_t ga = gl + off;
      asm volatile("global_load_async_to_lds_b128 %0, %1, off"
                   :: "v"(la), "v"(ga) : "memory");
    }
#endif
    asm volatile("s_wait_asynccnt 0" ::: "memory");
#if HAVE_TDM
    __builtin_amdgcn_s_wait_tensorcnt(0);
#endif
  }
#else
  for (int idx = tid; idx < Q_N * 3; idx += NT) s_rawlog[idx] = lg[idx];
  for (int idx = tid; idx < 288; idx += NT)     s_rawlab[idx] = lb[idx];
#endif
  __syncthreads();

  // -------- unpack: {bx, by, sigmoid} packed per query; target bboxes --------
  for (int q = tid; q < Q_N; q += NT) {
    float l0 = s_rawlog[3 * q];
    s_qd[q] = make_float4(s_rawlog[3 * q + 1],
                          s_rawlog[3 * q + 2],
                          1.0f / (1.0f + __expf(-l0)),
                          0.0f);
  }
  if (tid < K_N) {
    s_tx[tid] = s_rawlab[3 * tid + 1];
    s_ty[tid] = s_rawlab[3 * tid + 2];
    s_a[tid]  = 0;
  }
  if (tid <= K_N) s_u[tid] = 0.0f;
  for (int j = tid; j <= Q_N; j += NT) {
    s_colv[j] = make_float4(INF_F, 0.0f, 0.0f, 0.0f);  // minv, way, used, v=0
    s_p[j] = 0;
  }
  __syncthreads();

  // -------- Jonker-Volgenant LAP with lazy potentials, state in LDS --------
  // Invariants vs. reference (D = cumulative delta of this row's path):
  //   unused j : A[j] = minv_ref[j] + D          (A = s_colv[j].x, absolute)
  //   used j   : A[j] = Dm[j] (D value when j joined the path)
  //              v_ref[j] = v[j] - (D - Dm[j]);  u_ref[p[j]] = u[p[j]] + (D - Dm[j])
  //   delta step: D_new = A[j1]  (the selected minimum itself)
  for (int i = 1; i <= K_N; ++i) {
    for (int j = tid; j <= Q_N; j += NT) { s_colv[j].x = INF_F; s_colv[j].z = 0.0f; }
    if (tid == 0) { s_p[0] = i; s_colv[0].x = 0.0f; s_colv[0].z = 1.0f; }  // Dm[0]=0
    __syncthreads();

    int   j0 = 0;
    int   j1 = 0;
    float dfin = 0.0f;                          // running D == A[j1] after each step
    while (true) {
      const int   i0  = s_p[j0];
      const float c0  = s_colv[j0].x - s_u[i0];   // Dm[j0] - u_stored[i0]
      const float txk = s_tx[i0 - 1];
      const float tyk = s_ty[i0 - 1];
      float best = INF_F;
      int   bestj = 1 << 30;
      #pragma unroll
      for (int c = 0; c < CPT; ++c) {
        const int j = tid + 1 + c * NT;
        float4 cs = s_colv[j];                  // one ds_load_b128: minv/way/used/v
        if (cs.z == 0.0f) {                     // unused
          float4 qd = s_qd[j - 1];              // one ds_load_b128: bx/by/sig
          // absolute reduced cost: cost(C^T[i0-1][q]) - u[i0] - v[j] + Dm[j0]
          float h = fabsf(qd.x - txk) + fabsf(qd.y - tyk) - qd.z - cs.w + c0;
          float m = cs.x;
          if (h < m) {
            m = h;                              // minv+way update: one ds_store_b64
            *reinterpret_cast<float2*>(&s_colv[j]) =
                make_float2(h, __int_as_float(j0));
          }
          argmin2(best, bestj, m, j);
        }
      }
      // wave32 shuffle argmin, then every thread combines the 8 wave partials
      for (int off = 16; off > 0; off >>= 1) {
        float ov = __shfl_down(best, (unsigned)off, 32);
        int   oj = __shfl_down(bestj, (unsigned)off, 32);
        argmin2(best, bestj, ov, oj);
      }
      if ((tid & 31) == 0) { s_redv[tid >> 5] = best; s_redj[tid >> 5] = bestj; }
      __syncthreads();
      float bv = s_redv[0]; int bj = s_redj[0];
      #pragma unroll
      for (int w = 1; w < NT / 32; ++w) argmin2(bv, bj, s_redv[w], s_redj[w]);
      j1 = bj;
      dfin = bv;                                // D_new = A[j1]
      if (tid == 0) s_colv[j1].z = 1.0f;        // mark used; A[j1] == Dm[j1] == dfin
      __syncthreads();
      if (s_p[j1] == 0) break;                  // uniform exit: free column found
      j0 = j1;
    }

    // row-end fixup: apply all deferred v/u updates in one pass.
    // (For the terminal free column dfin - A[j1] == 0, so including it is a no-op.)
    #pragma unroll
    for (int c = 0; c < CPT; ++c) {
      const int j = tid + 1 + c * NT;
      float4 cs = s_colv[j];
      if (cs.z != 0.0f) {                       // used
        float dj = dfin - cs.x;                 // D_final - Dm[j]
        s_colv[j].w = cs.w - dj;                // v[j] -= dj
        atomicAdd(&s_u[s_p[j]], dj);            // ds_add_f32 scatter (rows distinct)
      }
    }
    if (tid == 0) {
      s_colv[0].w -= dfin;                      // v[0] -= D_final
      atomicAdd(&s_u[s_p[0]], dfin);            // u[i] accrues every delta (column 0)
    }
    __syncthreads();

    // augment along the alternating path (sequential, tiny)
    if (tid == 0) {
      int jj = j1;
      do {
        int jn = __float_as_int(s_colv[jj].y);  // way[jj]
        s_p[jj] = s_p[jn];
        jj = jn;
      } while (jj != 0);
    }
    __syncthreads();
  }

  // -------- a[row] = assigned column; then rank-sort 96 values --------
  for (int j = tid + 1; j <= Q_N; j += NT)
    if (s_p[j] > 0) s_a[s_p[j] - 1] = j - 1;
  __syncthreads();
  if (tid < K_N) {
    const int val = s_a[tid];
    int rank = 0;
    for (int s = 0; s < K_N; ++s) rank += (s_a[s] < val) ? 1 : 0;
    out[(size_t)b * K_N + rank]                          = (float)val;  // row_ind = sort(a)
    out[(size_t)BS_N * K_N + (size_t)b * K_N + rank]     = (float)tid;  // col_ind = argsort(a)
  }
}

extern "C" void kernel_launch(void* const* d_in, const int* in_sizes, int n_in,
                              void* d_out, int out_size, void* d_ws, size_t ws_size,
                              hipStream_t stream) {
  (void)in_sizes; (void)n_in; (void)d_ws; (void)ws_size; (void)out_size;
  const float* logits = (const float*)d_in[0];
  const float* labels = (const float*)d_in[1];
  float* out = (float*)d_out;
  hungarian_kernel<<<dim3(BS_N), dim3(NT), 0, stream>>>(logits, labels, out);
}